// FullAreaUNet_31233002177257
// MI455X (gfx1250) — compile-verified
//
#include <hip/hip_runtime.h>
#include <cstddef>
#include <cstdint>

#define NF_N 150000
#define NM_N 22000
#define NC_N 3300
#define NCONV 10
#define D_IN 208
#define DF 64
#define DM 128
#define DC 256

typedef __attribute__((ext_vector_type(2)))  float  v2f;
typedef __attribute__((ext_vector_type(8)))  float  v8f;

#ifndef __has_builtin
#define __has_builtin(x) 0
#endif

// ---------------------------------------------------------------------------
// Device helpers
// ---------------------------------------------------------------------------
__device__ inline float wave_sum(float v) {
  v += __shfl_xor(v, 16, 32);
  v += __shfl_xor(v,  8, 32);
  v += __shfl_xor(v,  4, 32);
  v += __shfl_xor(v,  2, 32);
  v += __shfl_xor(v,  1, 32);
  return v;
}

__device__ inline float gelu_exact(float x) {
  return 0.5f * x * (1.0f + erff(x * 0.70710678118654752f));
}

// ---------------------------------------------------------------------------
// Elementwise / graph kernels
// ---------------------------------------------------------------------------
__global__ void k_fill(float* __restrict__ p, size_t n, float v) {
  size_t i = (size_t)blockIdx.x * blockDim.x + threadIdx.x;
  if (i < n) p[i] = v;
}

__global__ void k_add(const float* __restrict__ a, const float* __restrict__ b,
                      float* __restrict__ y, size_t n) {
  size_t i = (size_t)blockIdx.x * blockDim.x + threadIdx.x;
  if (i < n) y[i] = a[i] + b[i];
}

__global__ void k_deg(const int* __restrict__ col, const float* __restrict__ w,
                      float* __restrict__ deg, int E) {
  int e = blockIdx.x * blockDim.x + threadIdx.x;
  if (e < E) atomicAdd(&deg[col[e]], w[e]);
}

__global__ void k_dinv(float* __restrict__ deg, int n) {
  int i = blockIdx.x * blockDim.x + threadIdx.x;
  if (i < n) {
    float d = deg[i];
    deg[i] = (d > 0.0f) ? rsqrtf(d) : 0.0f;
  }
}

__global__ void k_edge_norm(const int* __restrict__ row, const int* __restrict__ col,
                            const float* __restrict__ w, const float* __restrict__ dinv,
                            float* __restrict__ norm, int E) {
  int e = blockIdx.x * blockDim.x + threadIdx.x;
  if (e < E) norm[e] = dinv[row[e]] * w[e] * dinv[col[e]];
}

// scat = dinv[row]^2 * h + bias[col]   (GCN self-loop + bias, pre-scatter init)
template <int D>
__global__ void k_scat_init(const float* __restrict__ h, const float* __restrict__ dinv,
                            const float* __restrict__ b, float* __restrict__ scat,
                            size_t total) {
  size_t i = (size_t)blockIdx.x * blockDim.x + threadIdx.x;
  if (i >= total) return;
  int row = (int)(i / D);          // D is a power of two -> shifts
  int col = (int)(i % D);
  float di = dinv[row];
  scat[i] = di * di * h[i] + b[col];
}

// out[dst[e], :] += val[e] * x[src[e], :]   thread = one edge x 4 columns.
// LOGG = log2(D/4) so index math is shift/mask.
template <int LOGG>
__global__ void k_scatter(const int* __restrict__ dst, const int* __restrict__ src,
                          const float* __restrict__ val, const float* __restrict__ x,
                          float* __restrict__ out, size_t total) {
  size_t i = (size_t)blockIdx.x * blockDim.x + threadIdx.x;
  if (i >= total) return;
  constexpr int D = 4 << LOGG;
  int e = (int)(i >> LOGG);
  int q = (int)(i & ((1u << LOGG) - 1)) << 2;
  int s = src[e], d = dst[e];
  float v = val[e];
  const float4 xv = *reinterpret_cast<const float4*>(x + (size_t)s * D + q);
  float* o = out + (size_t)d * D + q;
  atomicAdd(o + 0, v * xv.x);
  atomicAdd(o + 1, v * xv.y);
  atomicAdd(o + 2, v * xv.z);
  atomicAdd(o + 3, v * xv.w);
}

// ---------------------------------------------------------------------------
// Wave-per-row layernorm (+ optional exact GELU)
// ---------------------------------------------------------------------------
__global__ void k_ln_act(const float* __restrict__ x, const float* __restrict__ g,
                         const float* __restrict__ bn, float* __restrict__ y,
                         int N, int D, int doGelu) {
  int wave = (int)(((size_t)blockIdx.x * blockDim.x + threadIdx.x) >> 5);
  int lane = threadIdx.x & 31;
  if (wave >= N) return;
  const float* xr = x + (size_t)wave * D;
  float vals[8];
  int cnt = (D + 31) >> 5;
  float s = 0.f;
  for (int i = 0; i < cnt; ++i) {
    int d = lane + (i << 5);
    float v = (d < D) ? xr[d] : 0.f;
    vals[i] = v; s += v;
  }
  s = wave_sum(s);
  float mean = s / (float)D;
  float vs = 0.f;
  for (int i = 0; i < cnt; ++i) {
    int d = lane + (i << 5);
    if (d < D) { float t = vals[i] - mean; vs += t * t; }
  }
  vs = wave_sum(vs);
  float rstd = rsqrtf(vs / (float)D + 1e-5f);
  float* yr = y + (size_t)wave * D;
  for (int i = 0; i < cnt; ++i) {
    int d = lane + (i << 5);
    if (d < D) {
      float t = (vals[i] - mean) * rstd * g[d] + bn[d];
      if (doGelu) t = gelu_exact(t);
      yr[d] = t;
    }
  }
}

__global__ void k_l2norm(const float* __restrict__ x, float* __restrict__ y, int N, int D) {
  int wave = (int)(((size_t)blockIdx.x * blockDim.x + threadIdx.x) >> 5);
  int lane = threadIdx.x & 31;
  if (wave >= N) return;
  const float* xr = x + (size_t)wave * D;
  float vals[8];
  int cnt = (D + 31) >> 5;
  float s = 0.f;
  for (int i = 0; i < cnt; ++i) {
    int d = lane + (i << 5);
    float v = (d < D) ? xr[d] : 0.f;
    vals[i] = v; s += v * v;
  }
  s = wave_sum(s);
  float inv = 1.0f / fmaxf(sqrtf(s), 1e-8f);
  float* yr = y + (size_t)wave * D;
  for (int i = 0; i < cnt; ++i) {
    int d = lane + (i << 5);
    if (d < D) yr[d] = vals[i] * inv;
  }
}

// ---------------------------------------------------------------------------
// WMMA GEMM: Y[N,M] = A[N,K] @ B[K,M] (+ bias). One wave = one 16x16 tile,
// 8 waves/block = 128 rows; grid.y = M/16 column tiles.
// The block's 16-column slice of B (K x 16) is staged transposed in LDS
// (stride K+2: keeps ds_load_b64 8B-aligned, avoids bank conflicts), so the
// inner loop is: 1 global b64 (A) + 1 ds b64 (B) + 1 v_wmma per 4 K.
// Requires K % 4 == 0 (true for all shapes in this model).
// ---------------------------------------------------------------------------
template <int SANITIZE>
__global__ __launch_bounds__(256)
void k_gemm(const float* __restrict__ A, const float* __restrict__ B,
            const float* __restrict__ bias, float* __restrict__ Y,
            int N, int K, int M, int hasBias) {
#if defined(__gfx1250__) && __has_builtin(__builtin_amdgcn_wmma_f32_16x16x4_f32)
  __shared__ float bs[16 * 258];             // up to K=256 (+2 pad) x 16 cols
  const int tid = threadIdx.x;
  const int c0 = blockIdx.y * 16;
  const int stride = K + 2;
  for (int idx = tid; idx < (K << 4); idx += 256) {
    int col = idx & 15;                      // coalesced: col fastest
    int k   = idx >> 4;
    bs[col * stride + k] = B[(size_t)k * M + (c0 + col)];
  }
  __syncthreads();

  const int wv   = tid >> 5;
  const int lane = tid & 31;
  const int r0 = blockIdx.x * 128 + wv * 16;
  if (r0 >= N) return;                       // wave-uniform: EXEC stays all-1s
  const int mrow = lane & 15;
  const int hi   = lane >> 4;                // 0: lanes 0-15, 1: lanes 16-31
  int arow = r0 + mrow; if (arow >= N) arow = N - 1;
  const float* Ar = A + (size_t)arow * K + 2 * hi;
  const float* Bl = bs + mrow * stride + 2 * hi;

  v8f acc = {0.f, 0.f, 0.f, 0.f, 0.f, 0.f, 0.f, 0.f};
#pragma unroll 4
  for (int k = 0; k < K; k += 4) {
    v2f av = *reinterpret_cast<const v2f*>(Ar + k);   // global_load_b64
    if (SANITIZE) {
      if (!(av.x == av.x)) av.x = 0.f;
      if (!(av.y == av.y)) av.y = 0.f;
    }
    v2f bv = *reinterpret_cast<const v2f*>(Bl + k);   // ds_load_b64
    acc = __builtin_amdgcn_wmma_f32_16x16x4_f32(false, av, false, bv,
                                                (short)0, acc, false, false);
  }

  for (int i = 0; i < 8; ++i) {
    int row = r0 + i + (hi ? 8 : 0);
    int col = c0 + mrow;
    if (row < N) {
      float v = acc[i];
      if (hasBias) v += bias[col];
      Y[(size_t)row * M + col] = v;
    }
  }
#else
  (void)A; (void)B; (void)bias; (void)Y; (void)N; (void)K; (void)M; (void)hasBias;
#endif
}

// ---------------------------------------------------------------------------
// Host-side orchestration
// ---------------------------------------------------------------------------
struct LinLN { const float *W, *b, *g, *bn; };
struct Gcn   { const float *convW, *convb, *lng, *lnb; LinLN res; bool hasDown; LinLN down; };
struct Graph { const int *row, *col; const float *norm, *dinv; int E; };
struct Temps { float *h, *scat, *ping, *pong, *ident, *comb; };

static inline int cdiv(size_t a, size_t b) { return (int)((a + b - 1) / b); }

static void L_fill(hipStream_t s, float* p, size_t n, float v) {
  k_fill<<<cdiv(n, 256), 256, 0, s>>>(p, n, v);
}
static void L_add(hipStream_t s, const float* a, const float* b, float* y, size_t n) {
  k_add<<<cdiv(n, 256), 256, 0, s>>>(a, b, y, n);
}
static void L_gemm(hipStream_t s, const float* A, const float* W, const float* b,
                   float* Y, int N, int K, int M, bool bias, bool sanitize) {
  dim3 grid(cdiv(N, 128), cdiv(M, 16));
  if (sanitize)
    k_gemm<1><<<grid, 256, 0, s>>>(A, W, b, Y, N, K, M, bias ? 1 : 0);
  else
    k_gemm<0><<<grid, 256, 0, s>>>(A, W, b, Y, N, K, M, bias ? 1 : 0);
}
static void L_ln(hipStream_t s, const float* x, const float* g, const float* bn,
                 float* y, int N, int D, bool gelu) {
  k_ln_act<<<cdiv((size_t)N * 32, 256), 256, 0, s>>>(x, g, bn, y, N, D, gelu ? 1 : 0);
}
static void L_l2norm(hipStream_t s, const float* x, float* y, int N, int D) {
  k_l2norm<<<cdiv((size_t)N * 32, 256), 256, 0, s>>>(x, y, N, D);
}
static void L_scat_init(hipStream_t s, const float* h, const float* dinv, const float* b,
                        float* scat, int N, int D) {
  size_t total = (size_t)N * D;
  int blocks = cdiv(total, 256);
  if (D == 64)       k_scat_init<64 ><<<blocks, 256, 0, s>>>(h, dinv, b, scat, total);
  else if (D == 128) k_scat_init<128><<<blocks, 256, 0, s>>>(h, dinv, b, scat, total);
  else               k_scat_init<256><<<blocks, 256, 0, s>>>(h, dinv, b, scat, total);
}
static void L_scatter(hipStream_t s, const int* dst, const int* src, const float* val,
                      const float* x, float* out, int nnz, int D) {
  size_t total = (size_t)nnz * (D >> 2);
  int blocks = cdiv(total, 256);
  if (D == 64)       k_scatter<4><<<blocks, 256, 0, s>>>(dst, src, val, x, out, total);
  else if (D == 128) k_scatter<5><<<blocks, 256, 0, s>>>(dst, src, val, x, out, total);
  else               k_scatter<6><<<blocks, 256, 0, s>>>(dst, src, val, x, out, total);
}

// out = result of 10 GCN convs starting from x (x not modified)
static const float* gcn_stack(hipStream_t s, const float* x, const Gcn& B, const Graph& G,
                              int N, int D, Temps& T) {
  const float* cur = x;
  float* bufs[2] = { T.ping, T.pong };
  int which = 0;
  for (int i = 0; i < NCONV; ++i) {
    L_gemm(s, cur, B.convW + (size_t)i * D * D, nullptr, T.h, N, D, D, false, false);
    L_scat_init(s, T.h, G.dinv, B.convb + (size_t)i * D, T.scat, N, D);
    L_scatter(s, G.col, G.row, G.norm, T.h, T.scat, G.E, D);
    L_ln(s, T.scat, B.lng + (size_t)i * D, B.lnb + (size_t)i * D, bufs[which], N, D, true);
    cur = bufs[which];
    which ^= 1;
  }
  return cur;
}

static void enc_block(hipStream_t s, const float* x, const Gcn& B, const Graph& G,
                      int N, int D, Temps& T, float* result) {
  L_gemm(s, x, B.res.W, B.res.b, T.h, N, D, D, true, false);
  L_ln(s, T.h, B.res.g, B.res.bn, T.ident, N, D, true);
  const float* out = gcn_stack(s, x, B, G, N, D, T);
  L_add(s, out, T.ident, result, (size_t)N * D);
}

static void dec_block(hipStream_t s, const float* x, const float* skip, const Gcn& B,
                      const Graph& G, int N, int D, int Dres, Temps& T, float* result) {
  L_add(s, x, skip, T.comb, (size_t)N * D);
  L_gemm(s, T.comb, B.res.W, B.res.b, T.h, N, D, Dres, true, false);
  L_ln(s, T.h, B.res.g, B.res.bn, T.ident, N, Dres, true);
  const float* out = gcn_stack(s, T.comb, B, G, N, D, T);
  if (B.hasDown) {
    L_gemm(s, out, B.down.W, B.down.b, T.h, N, D, Dres, true, false);
    L_ln(s, T.h, B.down.g, B.down.bn, T.scat, N, Dres, false);
    out = T.scat;
  }
  L_add(s, out, T.ident, result, (size_t)N * Dres);
}

static void sparse_map(hipStream_t s, const float* x, const int* rows, const int* cols,
                       const float* vals, int nnz, int Nout, int Din, int Dout,
                       const LinLN& P, Temps& T, float* result) {
  L_fill(s, T.ping, (size_t)Nout * Din, 0.f);
  L_scatter(s, rows, cols, vals, x, T.ping, nnz, Din);
  L_gemm(s, T.ping, P.W, P.b, T.h, Nout, Din, Dout, true, false);
  L_ln(s, T.h, P.g, P.bn, result, Nout, Dout, true);
}

extern "C" void kernel_launch(void* const* d_in, const int* in_sizes, int n_in,
                              void* d_out, int out_size, void* d_ws, size_t ws_size,
                              hipStream_t stream) {
  hipStream_t s = stream;
  const float* features = (const float*)d_in[0];
  const int*   ei_f = (const int*)d_in[1];   int EF = in_sizes[2];
  const float* ew_f = (const float*)d_in[2];
  const int*   ei_m = (const int*)d_in[3];   int EM = in_sizes[4];
  const float* ew_m = (const float*)d_in[4];
  const int*   ei_c = (const int*)d_in[5];   int EC = in_sizes[6];
  const float* ew_c = (const float*)d_in[6];
  const int*   f2m_rows = (const int*)d_in[7];
  const int*   f2m_cols = (const int*)d_in[8];
  const float* f2m_vals = (const float*)d_in[9];
  const int*   m2c_rows = (const int*)d_in[10];
  const int*   m2c_cols = (const int*)d_in[11];
  const float* m2c_vals = (const float*)d_in[12];
  const int*   c2m_rows = (const int*)d_in[13];
  const int*   c2m_cols = (const int*)d_in[14];
  const float* c2m_vals = (const float*)d_in[15];
  const int*   m2f_rows = (const int*)d_in[16];
  const int*   m2f_cols = (const int*)d_in[17];
  const float* m2f_vals = (const float*)d_in[18];

  // ---- params: leaves in make_params insertion order starting at d_in[19]
  int pi = 19;
  auto getf = [&]() -> const float* { return (const float*)d_in[pi++]; };
  auto getLin = [&]() -> LinLN { LinLN l; l.W = getf(); l.b = getf(); l.g = getf(); l.bn = getf(); return l; };
  auto getGcn = [&](bool down) -> Gcn {
    Gcn g; g.convW = getf(); g.convb = getf(); g.lng = getf(); g.lnb = getf();
    g.res = getLin(); g.hasDown = down;
    if (down) g.down = getLin(); else g.down = g.res;
    return g;
  };
  LinLN fusion = getLin();
  Gcn enc1 = getGcn(false), enc2 = getGcn(false), enc3 = getGcn(false);
  LinLN map_f2m = getLin(), map_m2c = getLin(), map_c2m = getLin(), map_m2f = getLin();
  Gcn dec3 = getGcn(true), dec2 = getGcn(true), dec1 = getGcn(false);
  LinLN out_fine = getLin(), out_mid = getLin(), out_coarse = getLin();
  LinLN recon1 = getLin(), recon2 = getLin();

  // ---- workspace bump allocator
  char* wp = (char*)d_ws;
  auto alloc = [&](size_t nfl) -> float* {
    float* r = (float*)wp;
    wp += ((nfl * sizeof(float) + 255) & ~(size_t)255);
    return r;
  };
  const size_t TS = (size_t)NF_N * DF;   // largest live N*D for any temp
  Temps T;
  T.h = alloc(TS); T.scat = alloc(TS); T.ping = alloc(TS);
  T.pong = alloc(TS); T.ident = alloc(TS); T.comb = alloc(TS);

  float* x0 = alloc((size_t)NF_N * DF);
  float* x1 = alloc((size_t)NF_N * DF);
  float* m1 = alloc((size_t)NM_N * DM);
  float* x2 = alloc((size_t)NM_N * DM);
  float* m2 = alloc((size_t)NC_N * DC);
  float* x3 = alloc((size_t)NC_N * DC);
  float* d3 = alloc((size_t)NC_N * DM);
  float* u2 = alloc((size_t)NM_N * DM);
  float* d2 = alloc((size_t)NM_N * DF);
  float* u1 = alloc((size_t)NF_N * DF);
  float* d1 = alloc((size_t)NF_N * DF);
  float* dinvF = alloc(NF_N); float* dinvM = alloc(NM_N); float* dinvC = alloc(NC_N);
  float* normF = alloc(EF);   float* normM = alloc(EM);   float* normC = alloc(EC);

  float* out = (float*)d_out;
  float* emb_fine   = out;
  float* emb_mid    = emb_fine + (size_t)NF_N * DF;
  float* emb_coarse = emb_mid + (size_t)NM_N * DF;
  float* recon      = emb_coarse + (size_t)NC_N * DF;

  // ---- GCN normalization (self-loop weight 1 folded into deg init; self-loop
  //      message dinv^2 * h and bias are applied by k_scat_init)
  auto build_graph = [&](const int* ei, const float* ew, int E, int n,
                         float* dinv, float* norm) -> Graph {
    L_fill(s, dinv, n, 1.0f);                               // self-loop weight
    k_deg<<<cdiv(E, 256), 256, 0, s>>>(ei + E, ew, dinv, E);
    k_dinv<<<cdiv(n, 256), 256, 0, s>>>(dinv, n);
    k_edge_norm<<<cdiv(E, 256), 256, 0, s>>>(ei, ei + E, ew, dinv, norm, E);
    Graph g; g.row = ei; g.col = ei + E; g.norm = norm; g.dinv = dinv; g.E = E;
    return g;
  };
  Graph gF = build_graph(ei_f, ew_f, EF, NF_N, dinvF, normF);
  Graph gM = build_graph(ei_m, ew_m, EM, NM_N, dinvM, normM);
  Graph gC = build_graph(ei_c, ew_c, EC, NC_N, dinvC, normC);

  // ---- fusion: x0 = l2norm(gelu(ln(nan_to_num(features) @ W + b)))
  L_gemm(s, features, fusion.W, fusion.b, T.h, NF_N, D_IN, DF, true, /*sanitize=*/true);
  L_ln(s, T.h, fusion.g, fusion.bn, T.ping, NF_N, DF, true);
  L_l2norm(s, T.ping, x0, NF_N, DF);

  // ---- encoder
  enc_block(s, x0, enc1, gF, NF_N, DF, T, x1);
  sparse_map(s, x1, f2m_rows, f2m_cols, f2m_vals, NF_N, NM_N, DF, DM, map_f2m, T, m1);
  enc_block(s, m1, enc2, gM, NM_N, DM, T, x2);
  sparse_map(s, x2, m2c_rows, m2c_cols, m2c_vals, NM_N, NC_N, DM, DC, map_m2c, T, m2);
  enc_block(s, m2, enc3, gC, NC_N, DC, T, x3);

  // ---- decoder
  dec_block(s, x3, m2, dec3, gC, NC_N, DC, DM, T, d3);
  sparse_map(s, d3, c2m_rows, c2m_cols, c2m_vals, NM_N, NM_N, DM, DM, map_c2m, T, u2);
  dec_block(s, u2, x2, dec2, gM, NM_N, DM, DF, T, d2);
  sparse_map(s, d2, m2f_rows, m2f_cols, m2f_vals, NF_N, NF_N, DF, DF, map_m2f, T, u1);
  dec_block(s, u1, x1, dec1, gF, NF_N, DF, DF, T, d1);

  // ---- output heads (mlp_ln, no activation)
  L_gemm(s, d1, out_fine.W, out_fine.b, T.h, NF_N, DF, DF, true, false);
  L_ln(s, T.h, out_fine.g, out_fine.bn, emb_fine, NF_N, DF, false);
  L_gemm(s, d2, out_mid.W, out_mid.b, T.h, NM_N, DF, DF, true, false);
  L_ln(s, T.h, out_mid.g, out_mid.bn, emb_mid, NM_N, DF, false);
  L_gemm(s, d3, out_coarse.W, out_coarse.b, T.h, NC_N, DM, DF, true, false);
  L_ln(s, T.h, out_coarse.g, out_coarse.bn, emb_coarse, NC_N, DF, false);

  // ---- reconstruction head (fine): mlp_ln(mlp_ln_gelu(emb_fine))
  L_gemm(s, emb_fine, recon1.W, recon1.b, T.h, NF_N, DF, DF, true, false);
  L_ln(s, T.h, recon1.g, recon1.bn, T.ping, NF_N, DF, true);
  L_gemm(s, T.ping, recon2.W, recon2.b, recon, NF_N, DF, D_IN, true, false);
  L_ln(s, recon, recon2.g, recon2.bn, recon, NF_N, D_IN, false);  // in-place: wave owns row

  (void)n_in; (void)out_size; (void)ws_size; (void)in_sizes;
}